// GNN_8967891714158
// MI455X (gfx1250) — compile-verified
//
#include <hip/hip_runtime.h>
#include <hip/hip_bf16.h>

// ---------------------------------------------------------------------------
// GIN forward for MI455X (gfx1250, wave32).
//  N=50000 nodes, E=600000 edges, D=128, L=5.
//  - gather+edge-proj+scatter fused, e recomputed per edge (saves ~1.2GB HBM)
//  - MLP GEMMs via V_WMMA_F32_16X16X4_F32 (full fp32 reference precision)
//  - BatchNorm via per-channel atomic partial sums + apply kernel
// ---------------------------------------------------------------------------

#define N_NODES 50000
#define N_EDGES 600000
#define DIM     128
#define DIM2    256
#define NLAYERS 5
#define BN_EPS  1e-5f

typedef float v2f __attribute__((ext_vector_type(2)));
typedef float v8f __attribute__((ext_vector_type(8)));

__device__ __forceinline__ void atomic_add_f32(float* p, float v) {
    // hardware GLOBAL_ATOMIC_ADD_F32 (no CAS loop)
    unsafeAtomicAdd(p, v);
}

// --------------------------------------------------------------------------
// zero-fill
// --------------------------------------------------------------------------
__global__ void k_zero(float* __restrict__ p, int n) {
    int i = blockIdx.x * blockDim.x + threadIdx.x;
    if (i < n) p[i] = 0.0f;
}

// --------------------------------------------------------------------------
// h = x @ Wx + bx     x:[N,78]  Wx:[78,128]  -> h:[N,128]
// one node per block, one channel per thread; x row staged in LDS.
// --------------------------------------------------------------------------
__global__ void k_node_proj(const float* __restrict__ x,
                            const float* __restrict__ Wx,
                            const float* __restrict__ bx,
                            float* __restrict__ h) {
    __shared__ float sx[78];
    const int n = blockIdx.x;
    const int c = threadIdx.x;            // 128 threads
    if (c < 78) sx[c] = x[n * 78 + c];
    __syncthreads();
    float acc = bx[c];
#pragma unroll
    for (int k = 0; k < 78; ++k)
        acc = fmaf(sx[k], Wx[k * DIM + c], acc);
    h[n * DIM + c] = acc;
}

// --------------------------------------------------------------------------
// fused GIN message + aggregate:
//   agg[dst[e]] += h[src[e]] + edge_attr[e] @ We + be
// 256 threads = 2 edge-lanes x 128 channels; We/be cached in LDS.
// --------------------------------------------------------------------------
#define EDGES_PER_BLOCK 64
__global__ void k_edge_msg(const float* __restrict__ h,
                           const float* __restrict__ edge_attr,
                           const float* __restrict__ We,
                           const float* __restrict__ be,
                           const int* __restrict__ src,
                           const int* __restrict__ dst,
                           float* __restrict__ agg) {
    __shared__ float sWe[7 * DIM];
    __shared__ float sbe[DIM];
    const int tid = threadIdx.x;          // 256
    for (int i = tid; i < 7 * DIM; i += 256) sWe[i] = We[i];
    if (tid < DIM) sbe[tid] = be[tid];
    __syncthreads();

    const int c    = tid & (DIM - 1);
    const int slot = tid >> 7;            // 0 or 1
    const int base = blockIdx.x * EDGES_PER_BLOCK;

    for (int i = slot; i < EDGES_PER_BLOCK; i += 2) {
        const int e = base + i;
        if (e >= N_EDGES) break;
        const int s = src[e];
        const int d = dst[e];
        float acc = h[s * DIM + c] + sbe[c];
#pragma unroll
        for (int k = 0; k < 7; ++k)
            acc = fmaf(edge_attr[e * 7 + k], sWe[k * DIM + c], acc);
        atomic_add_f32(&agg[d * DIM + c], acc);
    }
}

// --------------------------------------------------------------------------
// C[M x NCOL] = act(A[M x K] @ B[K x NCOL] + bias)   via V_WMMA_F32_16X16X4_F32
// block = 128 threads (4 waves). Each wave owns one 16x16 output tile.
// A tile (16 x K) staged in LDS with padded stride; B served from L2.
// grid.x = M/16 row tiles, grid.y * 4 waves = NCOL/16 col tiles.
// --------------------------------------------------------------------------
template <int K, int NCOL, bool RELU>
__global__ void k_wmma_gemm(const float* __restrict__ A,
                            const float* __restrict__ B,
                            const float* __restrict__ bias,
                            float* __restrict__ C) {
    constexpr int LDA = K + 1;            // pad to avoid LDS bank conflicts
    __shared__ float sA[16 * LDA];

    const int tid  = threadIdx.x;         // 128
    const int lane = tid & 31;
    const int wave = tid >> 5;            // 0..3
    const int row0 = blockIdx.x * 16;
    const int col0 = (blockIdx.y * 4 + wave) * 16;

    // cooperative load of the 16 x K A tile
    for (int i = tid; i < 16 * K; i += 128) {
        const int r = i / K;
        const int k = i - r * K;
        sA[r * LDA + k] = A[(row0 + r) * K + k];
    }
    __syncthreads();

    const int r  = lane & 15;             // A: row within tile / B: col within tile
    const int hi = lane >> 4;             // 0: K pair {0,1}, 1: K pair {2,3}

    v8f acc = {};
#pragma unroll
    for (int kk = 0; kk < K; kk += 4) {
        v2f a, b;
        a.x = sA[r * LDA + kk + 2 * hi + 0];
        a.y = sA[r * LDA + kk + 2 * hi + 1];
        b.x = B[(kk + 2 * hi + 0) * NCOL + col0 + r];
        b.y = B[(kk + 2 * hi + 1) * NCOL + col0 + r];
        acc = __builtin_amdgcn_wmma_f32_16x16x4_f32(
            /*neg_a=*/false, a, /*neg_b=*/false, b,
            /*c_mod=*/(short)0, acc, /*reuse_a=*/false, /*reuse_b=*/false);
    }

    const float bv = bias[col0 + r];
#pragma unroll
    for (int i = 0; i < 8; ++i) {
        const int row = row0 + i + 8 * hi;
        const int col = col0 + r;
        float v = acc[i] + bv;
        if (RELU) v = fmaxf(v, 0.0f);
        C[row * NCOL + col] = v;
    }
}

// --------------------------------------------------------------------------
// BatchNorm statistics: stats[c] = sum z[:,c], stats[128+c] = sum z[:,c]^2
// 128 threads = 1 channel each; ROWS rows per block; atomic partial sums.
// --------------------------------------------------------------------------
#define BN_ROWS 128
__global__ void k_bn_stats(const float* __restrict__ z,
                           float* __restrict__ stats) {
    const int c  = threadIdx.x;           // 128
    const int r0 = blockIdx.x * BN_ROWS;
    float s = 0.0f, s2 = 0.0f;
    for (int i = 0; i < BN_ROWS; ++i) {
        const int r = r0 + i;
        if (r >= N_NODES) break;
        const float v = z[r * DIM + c];
        s  += v;
        s2 += v * v;
    }
    atomic_add_f32(&stats[c], s);
    atomic_add_f32(&stats[DIM + c], s2);
}

// --------------------------------------------------------------------------
// BatchNorm apply (+ optional ReLU):
//   out = gamma*(z - mu)*rsqrt(var + eps) + beta
// --------------------------------------------------------------------------
#define BN_APPLY_ROWS 8
__global__ void k_bn_apply(const float* __restrict__ z,
                           const float* __restrict__ stats,
                           const float* __restrict__ gamma,
                           const float* __restrict__ beta,
                           float* __restrict__ out,
                           int relu) {
    const int c  = threadIdx.x;           // 128
    const int r0 = blockIdx.x * BN_APPLY_ROWS;
    const float invN = 1.0f / (float)N_NODES;
    const float mu   = stats[c] * invN;
    const float var  = stats[DIM + c] * invN - mu * mu;
    const float inv  = rsqrtf(var + BN_EPS);
    const float g    = gamma[c];
    const float bt   = beta[c];
#pragma unroll
    for (int i = 0; i < BN_APPLY_ROWS; ++i) {
        const int r = r0 + i;
        float v = (z[r * DIM + c] - mu) * inv * g + bt;
        if (relu) v = fmaxf(v, 0.0f);
        out[r * DIM + c] = v;
    }
}

// --------------------------------------------------------------------------
// host-side orchestration
// --------------------------------------------------------------------------
extern "C" void kernel_launch(void* const* d_in, const int* in_sizes, int n_in,
                              void* d_out, int out_size, void* d_ws, size_t ws_size,
                              hipStream_t stream) {
    const float* x         = (const float*)d_in[0];
    const float* edge_attr = (const float*)d_in[1];
    const float* Wx        = (const float*)d_in[2];
    const float* bx        = (const float*)d_in[3];
    const float* We        = (const float*)d_in[4];
    const float* be        = (const float*)d_in[5];
    const float* W1        = (const float*)d_in[6];   // [L,128,256]
    const float* b1        = (const float*)d_in[7];   // [L,256]
    const float* W2        = (const float*)d_in[8];   // [L,256,128]
    const float* b2        = (const float*)d_in[9];   // [L,128]
    const float* gamma     = (const float*)d_in[10];  // [L,128]
    const float* beta      = (const float*)d_in[11];  // [L,128]
    const int*   eidx      = (const int*)d_in[12];    // [2,E]
    const int*   src       = eidx;
    const int*   dst       = eidx + N_EDGES;

    float* ws    = (float*)d_ws;
    float* h     = ws;                         // N*128
    float* agg   = h + (size_t)N_NODES * DIM;  // N*128
    float* t     = agg + (size_t)N_NODES * DIM;    // N*256
    float* z     = t + (size_t)N_NODES * DIM2;     // N*128
    float* stats = z + (size_t)N_NODES * DIM;      // 256

    // input projection: h = x @ Wx + bx
    k_node_proj<<<N_NODES, 128, 0, stream>>>(x, Wx, bx, h);

    const int aggElems = N_NODES * DIM;
    const int edgeGrid = (N_EDGES + EDGES_PER_BLOCK - 1) / EDGES_PER_BLOCK;
    const int rowTiles = N_NODES / 16;         // 3125, exact

    for (int l = 0; l < NLAYERS; ++l) {
        // clear accumulators
        k_zero<<<(aggElems + 255) / 256, 256, 0, stream>>>(agg, aggElems);
        k_zero<<<1, 256, 0, stream>>>(stats, 2 * DIM);

        // fused gather + edge projection + scatter-add
        k_edge_msg<<<edgeGrid, 256, 0, stream>>>(h, edge_attr, We, be, src, dst, agg);

        // t = relu(agg @ W1[l] + b1[l])      [N,128]@[128,256]
        k_wmma_gemm<DIM, DIM2, true><<<dim3(rowTiles, DIM2 / 16 / 4), 128, 0, stream>>>(
            agg, W1 + (size_t)l * DIM * DIM2, b1 + (size_t)l * DIM2, t);

        // z = t @ W2[l] + b2[l]              [N,256]@[256,128]
        k_wmma_gemm<DIM2, DIM, false><<<dim3(rowTiles, DIM / 16 / 4), 128, 0, stream>>>(
            t, W2 + (size_t)l * DIM2 * DIM, b2 + (size_t)l * DIM, z);

        // batch-norm statistics + apply (ReLU on all but last layer)
        k_bn_stats<<<(N_NODES + BN_ROWS - 1) / BN_ROWS, 128, 0, stream>>>(z, stats);

        float* outp = (l == NLAYERS - 1) ? (float*)d_out : h;
        k_bn_apply<<<N_NODES / BN_APPLY_ROWS, 128, 0, stream>>>(
            z, stats, gamma + (size_t)l * DIM, beta + (size_t)l * DIM, outp,
            (l < NLAYERS - 1) ? 1 : 0);
    }
}